// PLEs_bad_64321430225119
// MI455X (gfx1250) — compile-verified
//
#include <hip/hip_runtime.h>
#include <hip/hip_bf16.h>

typedef float v2f __attribute__((ext_vector_type(2)));
typedef float v8f __attribute__((ext_vector_type(8)));

#define CIN   64
#define COUT  64
#define KPTS  4
#define HH    512
#define WW    512
#define NPIX  (HH * WW)          // 262144
// workspace layout (floats): proj[64][N] | flows[8][N] | logits[4][N]
#define WS_FLOWS  ((size_t)64 * NPIX)
#define WS_LOGITS ((size_t)72 * NPIX)

// ---------------------------------------------------------------------------
// Kernel 1: fused GEMM  D[80 x N] = Wcomb[80 x 64] * X[64 x N]  (+ bias)
// Rows 0..63 = proj (conv_w), 64..71 = flows (offset_w), 72..75 = logits
// (wconv_w), 76..79 = zero padding.  Uses V_WMMA_F32_16X16X4_F32 (full f32).
// Block = 320 threads = 10 waves; wave w -> (mtile = w%5, 16-pixel col block
// nsub = w/5); block covers 32 pixels.
// ---------------------------------------------------------------------------
__global__ __launch_bounds__(320)
void defconv_gemm_kernel(const float* __restrict__ input,     // [64][N]
                         const float* __restrict__ conv_w,    // [64][64]
                         const float* __restrict__ conv_b,    // [64]
                         const float* __restrict__ offset_w,  // [8][64]
                         const float* __restrict__ offset_b,  // [8]
                         const float* __restrict__ wconv_w,   // [4][64]
                         const float* __restrict__ wconv_b,   // [4]
                         float* __restrict__ ws)
{
    const int wave  = (int)threadIdx.x >> 5;   // 0..9
    const int lane  = (int)threadIdx.x & 31;
    const int mtile = wave % 5;                // 0..4
    const int nsub  = wave / 5;                // 0..1
    const int n0    = (int)blockIdx.x * 32 + nsub * 16;

    const int col   = lane & 15;               // N within 16-wide tile
    const int half  = lane >> 4;               // lane half selects K 0/1 vs 2/3
    const int kbase = half * 2;

    // ---- load A tile (16 rows x 64 cols, this lane holds row=mtile*16+col,
    //      columns kbase + 4s + {0,1}) into 32 VGPRs ----------------------
    const int row = mtile * 16 + col;          // combined-matrix row for A load
    const float* wrow;
    float rowscale = 1.0f;
    if (row < 64)       wrow = conv_w   + row * 64;
    else if (row < 72)  wrow = offset_w + (row - 64) * 64;
    else if (row < 76)  wrow = wconv_w  + (row - 72) * 64;
    else              { wrow = conv_w; rowscale = 0.0f; }   // zero pad rows

    float a[32];
#pragma unroll
    for (int s = 0; s < 16; ++s) {
        a[2 * s + 0] = rowscale * wrow[4 * s + kbase + 0];
        a[2 * s + 1] = rowscale * wrow[4 * s + kbase + 1];
    }

    // ---- K loop: 16 x wmma_f32_16x16x4_f32 ------------------------------
    v8f acc = {};
#pragma unroll
    for (int s = 0; s < 16; ++s) {
        const int c0 = 4 * s + kbase;
        v2f bfrag;
        bfrag.x = input[(size_t)(c0 + 0) * NPIX + n0 + col];
        bfrag.y = input[(size_t)(c0 + 1) * NPIX + n0 + col];
        v2f afrag;
        afrag.x = a[2 * s + 0];
        afrag.y = a[2 * s + 1];
        acc = __builtin_amdgcn_wmma_f32_16x16x4_f32(
            /*neg_a=*/false, afrag, /*neg_b=*/false, bfrag,
            /*c_mod=*/(short)0, acc, /*reuse_a=*/false, /*reuse_b=*/false);
    }

    // ---- writeback with bias; D layout: VGPR r -> row mtile*16 + r + 8*half
    const size_t n = (size_t)(n0 + col);
#pragma unroll
    for (int r = 0; r < 8; ++r) {
        const int ro = mtile * 16 + r + 8 * half;
        if (ro < 64) {
            ws[(size_t)ro * NPIX + n] = acc[r] + conv_b[ro];
        } else if (ro < 72) {
            ws[(size_t)ro * NPIX + n] = acc[r] + offset_b[ro - 64];
        } else if (ro < 76) {
            ws[(size_t)ro * NPIX + n] = acc[r] + wconv_b[ro - 72];
        }
        // rows 76..79: padding, dropped
    }
}

// ---------------------------------------------------------------------------
// Kernel 2: per-pixel softmax + flow gather + bilinear sample of proj + blend
// ---------------------------------------------------------------------------
__global__ __launch_bounds__(256)
void defconv_gather_kernel(const float* __restrict__ ax_p,
                           const float* __restrict__ ay_p,
                           const float* __restrict__ ws,
                           float* __restrict__ out)
{
    const int n = (int)blockIdx.x * 256 + (int)threadIdx.x;
    if (n >= NPIX) return;

    const float* __restrict__ proj   = ws;
    const float* __restrict__ flows  = ws + WS_FLOWS;
    const float* __restrict__ logits = ws + WS_LOGITS;

    const float ax = ax_p[n];
    const float ay = ay_p[n];
    const int ix = (int)ax;                 // trunc, coords are non-negative
    const int iy = (int)ay;
    const int m  = iy * WW + ix;            // gather point for flows

    // softmax over the 4 logits at this pixel
    float lg[KPTS], ek[KPTS];
#pragma unroll
    for (int k = 0; k < KPTS; ++k) lg[k] = logits[(size_t)k * NPIX + n];
    float mx = fmaxf(fmaxf(lg[0], lg[1]), fmaxf(lg[2], lg[3]));
    float ssum = 0.0f;
#pragma unroll
    for (int k = 0; k < KPTS; ++k) { ek[k] = __expf(lg[k] - mx); ssum += ek[k]; }
    const float inv_ssum = 1.0f / ssum;

    // fold softmax * bilinear * validity into 16 (weight, index) pairs
    float wgt16[16];
    int   idx16[16];
#pragma unroll
    for (int k = 0; k < KPTS; ++k) {
        const float wk = ek[k] * inv_ssum;
        const float fx = flows[(size_t)(2 * k + 0) * NPIX + m];
        const float fy = flows[(size_t)(2 * k + 1) * NPIX + m];
        const float xs = (ax + fx) * ((float)WW / (float)(WW - 1)) - 0.5f;
        const float ys = (ay + fy) * ((float)HH / (float)(HH - 1)) - 0.5f;
        const float x0f = floorf(xs);
        const float y0f = floorf(ys);
        const float wx = xs - x0f;
        const float wy = ys - y0f;
        const int x0 = (int)x0f;
        const int y0 = (int)y0f;
#pragma unroll
        for (int cy = 0; cy < 2; ++cy) {
#pragma unroll
            for (int cx = 0; cx < 2; ++cx) {
                const int xc = x0 + cx;
                const int yc = y0 + cy;
                const float cw = (cx ? wx : 1.0f - wx) * (cy ? wy : 1.0f - wy);
                const bool valid = (xc >= 0) & (xc < WW) & (yc >= 0) & (yc < HH);
                const int xcc = min(max(xc, 0), WW - 1);
                const int ycc = min(max(yc, 0), HH - 1);
                wgt16[k * 4 + cy * 2 + cx] = valid ? (wk * cw) : 0.0f;
                idx16[k * 4 + cy * 2 + cx] = ycc * WW + xcc;
            }
        }
    }

    // per channel: 16 gathered FMAs, one coalesced store
#pragma unroll 4
    for (int c = 0; c < COUT; ++c) {
        const float* __restrict__ p = proj + (size_t)c * NPIX;
        float s = 0.0f;
#pragma unroll
        for (int i = 0; i < 16; ++i) s = fmaf(wgt16[i], p[idx16[i]], s);
        out[(size_t)c * NPIX + n] = s;
    }
}

// ---------------------------------------------------------------------------
extern "C" void kernel_launch(void* const* d_in, const int* in_sizes, int n_in,
                              void* d_out, int out_size, void* d_ws, size_t ws_size,
                              hipStream_t stream)
{
    // setup_inputs order:
    // 0 input [1,64,512,512] f32     1 ax [1,N] f32        2 ay [1,N] f32
    // 3 conv_w [64,64] f32           4 conv_b [64] f32
    // 5 offset_w [4,2,64] f32        6 offset_b [4,2] f32
    // 7 wconv_w [4,64] f32           8 wconv_b [4] f32
    const float* input    = (const float*)d_in[0];
    const float* ax       = (const float*)d_in[1];
    const float* ay       = (const float*)d_in[2];
    const float* conv_w   = (const float*)d_in[3];
    const float* conv_b   = (const float*)d_in[4];
    const float* offset_w = (const float*)d_in[5];
    const float* offset_b = (const float*)d_in[6];
    const float* wconv_w  = (const float*)d_in[7];
    const float* wconv_b  = (const float*)d_in[8];
    float* out = (float*)d_out;
    float* ws  = (float*)d_ws;      // needs 76 * N * 4 bytes (~80 MB)

    (void)in_sizes; (void)n_in; (void)out_size; (void)ws_size;

    // Kernel 1: 32 pixels / block, 10 waves (5 m-tiles x 2 column blocks)
    defconv_gemm_kernel<<<NPIX / 32, 320, 0, stream>>>(
        input, conv_w, conv_b, offset_w, offset_b, wconv_w, wconv_b, ws);

    // Kernel 2: 1 thread per pixel
    defconv_gather_kernel<<<NPIX / 256, 256, 0, stream>>>(ax, ay, ws, out);
}